// LRNN_29308856828657
// MI455X (gfx1250) — compile-verified
//
#include <hip/hip_runtime.h>
#include <cstdint>
#include <cstddef>

// LRNN gated scan along W for layout (B,H,W,C)=(8,256,256,64) fp32.
// h_w = g_w*(h_{w-1} - x_w) + x_w, h_0 = 0, emit h at every w.
// Memory-bound (384 MiB @ 23.3 TB/s ~= 17 us). Strategy: per-thread serial scan
// over W with deep async prefetch (global_load_async_to_lds, triple-buffered,
// non-temporal) to keep ~24 MB in flight; coalesced b128 on the contiguous C
// axis; non-temporal b128 stores. All streams are read/write-once, so NT
// everywhere keeps the 192 MB L2 from being rinsed by 384 MiB of dead lines.

typedef float v4f __attribute__((ext_vector_type(4)));

#define WAIT_ASYNC(n) asm volatile("s_wait_asynccnt " #n ::: "memory")
#define WAIT_DS0()    asm volatile("s_wait_dscnt 0" ::: "memory")

constexpr int Bc = 8, Hc = 256, Wc = 256, Cc = 64;
constexpr int ROWS = 4;            // (b,h) rows per block
constexpr int TPR  = 16;           // threads per row: 16 x float4 covers C=64
constexpr int NT   = ROWS * TPR;   // 64 threads = 2 wave32
constexpr int T    = 8;            // w-steps per chunk
constexpr int NBUF = 3;            // triple buffer -> 48 outstanding b128/thread
constexpr int NCHUNK = Wc / T;     // 32
constexpr int XELEMS = NBUF * T * NT;        // v4f elements in X region
constexpr unsigned GREG_BYTES = XELEMS * 16; // byte offset of G region in LDS

__device__ __forceinline__ void issue_chunk(const float* __restrict__ Xp,
                                            const float* __restrict__ Gp,
                                            unsigned gbase, unsigned lds0,
                                            int chunk, int buf) {
#pragma unroll
  for (int j = 0; j < T; ++j) {
    // global byte offset of this thread's 4 channels at w = chunk*T + j
    unsigned voff = gbase + (unsigned)(chunk * T + j) * (Cc * 4);
    // LDS dest byte offsets (per-lane VGPR address for the async copy)
    unsigned ldsX = lds0 + (unsigned)((buf * T + j) * NT) * 16u;
    unsigned ldsG = ldsX + GREG_BYTES;
    // Read-once streams: non-temporal so they don't displace L2 residents.
    asm volatile("global_load_async_to_lds_b128 %0, %1, %2 th:TH_LOAD_NT"
                 :: "v"(ldsX), "v"(voff), "s"(Xp) : "memory");
    asm volatile("global_load_async_to_lds_b128 %0, %1, %2 th:TH_LOAD_NT"
                 :: "v"(ldsG), "v"(voff), "s"(Gp) : "memory");
  }
}

__global__ __launch_bounds__(NT) void lrnn_scan_kernel(
    const float* __restrict__ X, const float* __restrict__ G,
    float* __restrict__ Out) {
  __shared__ v4f smem[XELEMS * 2];  // [X: NBUF*T*NT][G: NBUF*T*NT] = 48 KB

  const int tid = threadIdx.x;
  const int r   = tid / TPR;        // row within block
  const int cg  = tid % TPR;        // channel group (4 channels)
  const int row = blockIdx.x * ROWS + r;   // (b,h) flat row, [0, 2048)

  // byte offset of (row, w=0, c=cg*4) in the (B,H,W,C) tensor
  const unsigned gbase = (unsigned)row * (unsigned)(Wc * Cc * 4)
                       + (unsigned)cg * 16u;
  // LDS byte offset of this thread's slot 0 (low 32 bits of flat LDS address
  // are the allocation-relative offset the async op's VGPR address expects)
  const unsigned lds0 = (unsigned)(uintptr_t)(&smem[0]) + (unsigned)tid * 16u;

  // Prologue: three chunks in flight (48 async b128 per thread outstanding,
  // ~24 MB in flight across the 1024 waves of the dispatch).
  issue_chunk(X, G, gbase, lds0, 0, 0);
  issue_chunk(X, G, gbase, lds0, 1, 1);
  issue_chunk(X, G, gbase, lds0, 2, 2);

  v4f h = {0.f, 0.f, 0.f, 0.f};
  const size_t obase = (size_t)row * (size_t)(Wc * Cc) + (size_t)cg * 4;

  for (int k = 0; k < NCHUNK; ++k) {
    const int buf = k % NBUF;

    // Async loads retire in issue order: chunk k is resident once at most
    // 16 * (#chunks issued beyond k) loads remain outstanding.
    if (k <= NCHUNK - 3)      WAIT_ASYNC(32);
    else if (k == NCHUNK - 2) WAIT_ASYNC(16);
    else                      WAIT_ASYNC(0);

#pragma unroll
    for (int j = 0; j < T; ++j) {
      const int idx = (buf * T + j) * NT + tid;
      v4f xv = smem[idx];             // ds_load_b128, conflict-free
      v4f gv = smem[XELEMS + idx];    // ds_load_b128
      h.x = __builtin_fmaf(gv.x, h.x - xv.x, xv.x);
      h.y = __builtin_fmaf(gv.y, h.y - xv.y, xv.y);
      h.z = __builtin_fmaf(gv.z, h.z - xv.z, xv.z);
      h.w = __builtin_fmaf(gv.w, h.w - xv.w, xv.w);
      // Streamed write-once output: non-temporal store.
      __builtin_nontemporal_store(
          h, (v4f*)(Out + obase + (size_t)(k * T + j) * Cc));
    }

    // Async LDS writes are unordered vs ds reads: drain DScnt before letting
    // the next chunk overwrite this buffer.
    WAIT_DS0();
    if (k + NBUF < NCHUNK)
      issue_chunk(X, G, gbase, lds0, k + NBUF, buf);
  }
}

extern "C" void kernel_launch(void* const* d_in, const int* in_sizes, int n_in,
                              void* d_out, int out_size, void* d_ws, size_t ws_size,
                              hipStream_t stream) {
  (void)in_sizes; (void)n_in; (void)out_size; (void)d_ws; (void)ws_size;
  const float* X = (const float*)d_in[0];
  const float* G = (const float*)d_in[1];
  float* Out = (float*)d_out;

  const int rows = Bc * Hc;                 // 2048 independent (b,h) rows
  dim3 grid(rows / ROWS);                   // 512 blocks
  dim3 block(NT);                           // 64 threads (2 wave32)
  lrnn_scan_kernel<<<grid, block, 0, stream>>>(X, G, Out);
}